// TopoHead_62474594287650
// MI455X (gfx1250) — compile-verified
//
#include <hip/hip_runtime.h>

// ---------------------------------------------------------------------------
// 0-dim sublevel-set persistence entropy, one 96x96 image per workgroup.
//   - TDM (tensor_load_to_lds) DMAs the image global->LDS      [CDNA5 path]
//   - 256 threads (8 wave32s) bitonic-sort the 9216 vertices by value in LDS
//   - lane 0 runs the elder-rule union-find sweep entirely in LDS
// LDS: 9216 f32 vals + 9216 i32 parent + 16384 u64 sort keys = 204800 B
// (fits the 320 KiB CDNA5 WGP LDS; impossible on CDNA4).
// ---------------------------------------------------------------------------

#define IMG_H    96
#define IMG_W    96
#define NPIX     (IMG_H * IMG_W)   // 9216
#define SORT_N   16384             // next pow2 >= NPIX
#define NTHREADS 256

typedef unsigned int       u32;
typedef unsigned long long u64;

typedef u32 u32x4 __attribute__((ext_vector_type(4)));
typedef int i32x4 __attribute__((ext_vector_type(4)));
typedef int i32x8 __attribute__((ext_vector_type(8)));

// Path-compressed find on the LDS parent array (serial phase only).
__device__ __forceinline__ int uf_find(int* p, int x) {
    int r = x;
    while (p[r] != r) r = p[r];
    while (p[x] != r) { int n = p[x]; p[x] = r; x = n; }
    return r;
}

__global__ __launch_bounds__(NTHREADS)
void topo_h0_entropy_kernel(const float* __restrict__ x, float* __restrict__ out) {
    extern __shared__ unsigned char smem[];
    float* vals   = (float*)smem;                        // [NPIX]   @ 0
    int*   parent = (int*)(smem + (size_t)NPIX * 4);     // [NPIX]   @ 36864
    u64*   keys   = (u64*)(smem + (size_t)NPIX * 8);     // [SORT_N] @ 73728 (8B aligned)

    const int tid = threadIdx.x;
    const int img = blockIdx.x;
    const unsigned long long gaddr =
        (unsigned long long)(const void*)(x + (size_t)img * NPIX);

    // ---- TDM: async DMA of the whole image (9216 f32) global -> LDS ----
    // D# group 0 (128b): count=1 | lds_addr=0 | global_addr[56:0] | type=2
    // D# group 1 (256b): data_size=4B, tensor_dim0=tile_dim0=9216, tile_dim1=1,
    //                    tensor_dim1=1, dim0_stride=dim1_stride=9216, no pad/iter.
    if (tid == 0) {
        u32x4 g0;
        g0[0] = 1u;                                            // count=1, user mode
        g0[1] = 0u;                                            // lds_addr = 0 (vals[])
        g0[2] = (u32)(gaddr & 0xFFFFFFFFull);                  // global_addr[31:0]
        g0[3] = (u32)((gaddr >> 32) & 0x1FFFFFFull)            // global_addr[56:32]
              | (2u << 30);                                    // type = 2 ("image")
        i32x8 g1;
        g1[0] = (int)(2u << 16);               // mask=0, data_size=2 (4 bytes)
        g1[1] = (int)((u32)NPIX << 16);        // tensor_dim0[15:0]=9216 (hi16 of word)
        g1[2] = (int)(1u << 16);               // tensor_dim0[31:16]=0, tensor_dim1=1
        g1[3] = (int)((u32)NPIX << 16);        // tensor_dim1 hi=0, tile_dim0=9216
        g1[4] = 1;                             // tile_dim1=1, tile_dim2=0
        g1[5] = NPIX;                          // tensor_dim0_stride[31:0]=9216
        g1[6] = (int)((u32)NPIX << 16);        // dim0_stride[47:32]=0, dim1_stride[15:0]=9216
        g1[7] = 0;                             // dim1_stride[47:16]=0
        i32x4 gz = {0, 0, 0, 0};               // groups 2/3 unused (<=2D tensor)
#if defined(__clang_major__) && __clang_major__ >= 23
        i32x8 gz8 = {0, 0, 0, 0, 0, 0, 0, 0};
        __builtin_amdgcn_tensor_load_to_lds(g0, g1, gz, gz, gz8, 0);
#else
        __builtin_amdgcn_tensor_load_to_lds(g0, g1, gz, gz, 0);
#endif
        __builtin_amdgcn_s_wait_tensorcnt(0);  // DMA complete before barrier
    }
    // Init union-find state (disjoint LDS region) while the DMA is in flight.
    for (int i = tid; i < NPIX; i += NTHREADS) parent[i] = -1;  // -1 => not yet born
    __syncthreads();

    // ---- Build sort keys: (orderable f32 bits << 32) | vertex index ----
    for (int i = tid; i < NPIX; i += NTHREADS) {
        u32 b = __float_as_uint(vals[i]);
        b ^= (b & 0x80000000u) ? 0xFFFFFFFFu : 0x80000000u;  // total order, ascending
        keys[i] = ((u64)b << 32) | (u32)i;
    }
    for (int i = NPIX + tid; i < SORT_N; i += NTHREADS) keys[i] = ~0ull;  // pad to end
    __syncthreads();

    // ---- Bitonic sort of 16384 u64 keys in LDS (all 8 waves) ----
    for (int k = 2; k <= SORT_N; k <<= 1) {
        for (int j = k >> 1; j > 0; j >>= 1) {
            for (int i = tid; i < SORT_N; i += NTHREADS) {
                int ixj = i ^ j;
                if (ixj > i) {
                    u64 a = keys[i], b = keys[ixj];
                    bool up = ((i & k) == 0);
                    if ((a > b) == up) { keys[i] = b; keys[ixj] = a; }
                }
            }
            __syncthreads();
        }
    }

    // ---- Elder-rule union-find sweep over vertices in filtration order ----
    // Adding vertex v at value dv activates edges to already-born neighbors,
    // each with weight max(vals) == dv. Root of a component == its min-value
    // vertex == birth vertex (elder rule keeps the older root). The resulting
    // (birth, death) multiset equals the reference's edge-sorted Kruskal.
    if (tid == 0) {
        double T = 0.0, S = 0.0;
        for (int k2 = 0; k2 < NPIX; ++k2) {
            int   v  = (int)(u32)keys[k2];
            float dv = vals[v];
            parent[v] = v;  // born: singleton component rooted at v
            int row = v / IMG_W;
            int col = v - row * IMG_W;
            int nb[4]; int nn = 0;
            if (col > 0)         nb[nn++] = v - 1;
            if (col < IMG_W - 1) nb[nn++] = v + 1;
            if (row > 0)         nb[nn++] = v - IMG_W;
            if (row < IMG_H - 1) nb[nn++] = v + IMG_W;
            for (int t = 0; t < nn; ++t) {
                int u = nb[t];
                if (parent[u] < 0) continue;            // neighbor not yet born
                int ru = uf_find(parent, u);
                int rv = uf_find(parent, v);
                if (ru == rv) continue;                 // same component: no pair
                float bu = vals[ru], bv = vals[rv];
                int younger = (bu >= bv) ? ru : rv;     // later birth dies
                int elder   = (bu >= bv) ? rv : ru;
                float pers  = dv - vals[younger];       // death - birth (>= 0)
                if (pers > 0.0f) {
                    double pd = (double)pers;
                    T += pd;
                    S += pd * log(pd);
                }
                parent[younger] = elder;
            }
        }
        // H = -sum p log p, p = pers/T  =>  H = log T - S/T  (0 if T == 0)
        float Hent = (T > 0.0) ? (float)(log(T) - S / T) : 0.0f;
        out[img] = Hent;
    }
}

extern "C" void kernel_launch(void* const* d_in, const int* in_sizes, int n_in,
                              void* d_out, int out_size, void* d_ws, size_t ws_size,
                              hipStream_t stream) {
    const float* x  = (const float*)d_in[0];
    float* out      = (float*)d_out;
    const int n_img = in_sizes[0] / NPIX;  // 8*8 = 64 independent images

    const size_t shmem = (size_t)NPIX * 4   // vals
                       + (size_t)NPIX * 4   // parent
                       + (size_t)SORT_N * 8; // sort keys  => 204800 B < 320 KiB WGP LDS

    topo_h0_entropy_kernel<<<dim3(n_img), dim3(NTHREADS), shmem, stream>>>(x, out);
}